// Im2RMMappingModule_56221121905135
// MI455X (gfx1250) — compile-verified
//
#include <hip/hip_runtime.h>
#include <math.h>

// ---------------- problem constants ----------------
#define BS      2
#define CFEA    64
#define HW      36864        // 192*192
#define NQ      812          // in-circle stereographic points (exact)
#define NPAD    832          // 52 tiles of 16
#define NTILES  52
#define COLS    80           // 64 fea + 3 sq + 3 key + 1 ones + 9 pad
#define COUT    69           // [rm_cos, var, qn(3), fea(64)]
#define HOWO    4096
#define KS      8            // k-split factor
#define KSLICE  (HW / KS)    // 4608
#define KB      128          // m-chunk staged in LDS
#define LSTR    (KB + 4)     // padded LDS row stride (floats), multiple of 4
#define BROWS   80           // uniform B panel rows in LDS
#define BDIM    256          // 8 waves per block

typedef __attribute__((ext_vector_type(2))) float v2f;
typedef __attribute__((ext_vector_type(8))) float v8f;

// ws float layout
//   qn   : [NPAD*3]            @ 0
//   idx  : [NPAD] (int)        @ 2496
//   dmax : [BS*NPAD]           @ 3328
//   S    : [BS*NPAD*COLS]      @ 4992
#define WS_QN    0
#define WS_IDX   2496
#define WS_DMAX  3328
#define WS_S     4992
#define ZERO_CNT (BS*NPAD + BS*NPAD*COLS)   // dmax..S contiguous

// ---------------- kernel 0: build stereographic grid + compacted indices ----
__global__ void k_grid(float* __restrict__ qn, int* __restrict__ idx) {
  if (threadIdx.x == 0 && blockIdx.x == 0) {
    int n = 0;
    for (int g = 0; g < HOWO; ++g) {
      int v = g >> 6, u = g & 63;
      float p = 4.0f * ((u + 0.5f) / 64.0f - 0.5f);
      float q = -4.0f * ((v + 0.5f) / 64.0f - 0.5f);
      float d = 1.0f + p * p + q * q;
      float z = -(-1.0f + p * p + q * q) / d;
      if (z > 0.0f) {
        qn[n * 3 + 0] = 2.0f * p / d;
        qn[n * 3 + 1] = 2.0f * q / d;
        qn[n * 3 + 2] = z;
        idx[n] = g;
        ++n;
      }
    }
    for (; n < NPAD; ++n) {   // zero-pad rows 812..831
      qn[n * 3 + 0] = 0.0f; qn[n * 3 + 1] = 0.0f; qn[n * 3 + 2] = 0.0f;
      idx[n] = 0;
    }
  }
}

// ---------------- zero fill ----------------
__global__ void k_zero(float* __restrict__ p, int n) {
  for (int i = blockIdx.x * blockDim.x + threadIdx.x; i < n;
       i += gridDim.x * blockDim.x)
    p[i] = 0.0f;
}

// ---------------- kernel 1: per-(b,n) max of cos over hw ----------------
__global__ void k_max(const float* __restrict__ key,   // [BS,3,HW]
                      const float* __restrict__ qn,
                      float* __restrict__ dmax) {
  const int n = blockIdx.x, b = blockIdx.y;
  const float qx = qn[n * 3 + 0], qy = qn[n * 3 + 1], qz = qn[n * 3 + 2];
  const float* kb = key + (size_t)b * 3 * HW;
  float mx = -1e30f;
  for (int m = threadIdx.x; m < HW; m += 256) {
    float c = qx * kb[m] + qy * kb[HW + m] + qz * kb[2 * HW + m];
    mx = fmaxf(mx, c);
  }
  __shared__ float red[256];
  red[threadIdx.x] = mx;
  __syncthreads();
  for (int s = 128; s > 0; s >>= 1) {
    if (threadIdx.x < (unsigned)s)
      red[threadIdx.x] = fmaxf(red[threadIdx.x], red[threadIdx.x + s]);
    __syncthreads();
  }
  if (threadIdx.x == 0) dmax[b * NPAD + n] = red[0];   // max cos (pre /sqrt3)
}

// ---------------- kernel 2: fused weights x [vals|sq|key|1] via f32 WMMA ----
// grid (NTILES, BS, KS), block 256 (8 waves; one 16-m subchunk per wave)
__global__ __launch_bounds__(BDIM)
void k_main(const float* __restrict__ fea,    // [BS,64,HW]
            const float* __restrict__ key,    // [BS,3,HW]
            const float* __restrict__ mask,   // [BS,1,HW]
            const float* __restrict__ lq,     // [1]
            const float* __restrict__ qn,
            const float* __restrict__ dmax,
            float* __restrict__ S) {          // [BS,NPAD,COLS] accumulators
  // Uniform B panel: rows 0-63 fea, 64-66 fea[61..63]^2, 67-69 key,
  // 70 ones, 71-79 zeros.  Reused as reduction scratch after the k-loop.
  __shared__ float lds_B[BROWS * LSTR];       // 80*132*4 = 42240 B
  __shared__ float lds_msk[LSTR];

  const int ntile = blockIdx.x, b = blockIdx.y, ks = blockIdx.z;
  const int tid  = threadIdx.x;
  const int lane = tid & 31, wave = tid >> 5;
  const int nrow = lane & 15, half = lane >> 4;

  const float scale = __expf(lq[0]) * 0.5773502691896258f;  // e^lq / sqrt(3)
  const int   n     = ntile * 16 + nrow;
  const float qx = qn[n * 3 + 0], qy = qn[n * 3 + 1], qz = qn[n * 3 + 2];
  const float dmx = dmax[b * NPAD + n];

  const float* feab = fea + (size_t)b * CFEA * HW;
  const float* keyb = key + (size_t)b * 3 * HW;
  const float* mskb = mask + (size_t)b * HW;

  // one-time chunk-invariant rows: 70 = ones, 71..79 = zeros
  for (int i = tid; i < 10 * KB; i += BDIM) {
    int row = 70 + i / KB, m = i % KB;
    lds_B[row * LSTR + m] = (row == 70) ? 1.0f : 0.0f;
  }

  v8f acc[5];
#pragma unroll
  for (int t = 0; t < 5; ++t) acc[t] = (v8f){0,0,0,0,0,0,0,0};

  const int mb   = wave * 16;       // this wave's 16-m subchunk within KB
  const int kbeg = ks * KSLICE;
  for (int k0 = kbeg; k0 < kbeg + KSLICE; k0 += KB) {
    __syncthreads();
    // ---- cooperative LDS staging (float4 along m) ----
    for (int i = tid; i < CFEA * (KB / 4); i += BDIM) {
      int row = i / (KB / 4), cv = (i % (KB / 4)) * 4;
      float4 v = *(const float4*)&feab[(size_t)row * HW + k0 + cv];
      *(float4*)&lds_B[row * LSTR + cv] = v;
      if (row >= 61) {              // squared fea rows 64..66 (no extra pass)
        float4 sq = make_float4(v.x * v.x, v.y * v.y, v.z * v.z, v.w * v.w);
        *(float4*)&lds_B[(row + 3) * LSTR + cv] = sq;
      }
    }
    for (int i = tid; i < 3 * (KB / 4); i += BDIM) {
      int row = i / (KB / 4), cv = (i % (KB / 4)) * 4;
      float4 v = *(const float4*)&keyb[(size_t)row * HW + k0 + cv];
      *(float4*)&lds_B[(67 + row) * LSTR + cv] = v;
    }
    if (tid < KB / 4)
      *(float4*)&lds_msk[tid * 4] = *(const float4*)&mskb[k0 + tid * 4];
    // ---- prefetch next chunk's fea (global_prefetch_b8 hint) ----
    if (k0 + KB < kbeg + KSLICE) {
      int prow = tid & 63, pseg = tid >> 6;            // 4 x 128B per row
      __builtin_prefetch(&feab[(size_t)prow * HW + k0 + KB + pseg * 32], 0, 1);
    }
    __syncthreads();

    // ---- 4 k-steps x 5 column tiles, fully uniform ----
#pragma unroll
    for (int kk = 0; kk < 4; ++kk) {
      const int mA = mb + kk * 4 + half * 2;   // this lane's m pair (even)
      v2f kx = *(const v2f*)&lds_B[67 * LSTR + mA];
      v2f ky = *(const v2f*)&lds_B[68 * LSTR + mA];
      v2f kz = *(const v2f*)&lds_B[69 * LSTR + mA];
      v2f mk = *(const v2f*)&lds_msk[mA];
      float c0 = qx * kx.x + qy * ky.x + qz * kz.x;
      float c1 = qx * kx.y + qy * ky.y + qz * kz.y;
      v2f a;
      a.x = __expf(scale * (c0 - dmx)) * mk.x;
      a.y = __expf(scale * (c1 - dmx)) * mk.y;
#pragma unroll
      for (int t = 0; t < 5; ++t) {
        v2f bv = *(const v2f*)&lds_B[(t * 16 + nrow) * LSTR + mA];
        acc[t] = __builtin_amdgcn_wmma_f32_16x16x4_f32(
            false, a, false, bv, (short)0, acc[t], false, false);
      }
    }
  }

  // ---- cross-wave reduction in LDS (reuse lds_B), then one atomic set ----
  __syncthreads();
#pragma unroll
  for (int t = 0; t < 5; ++t) {
#pragma unroll
    for (int r = 0; r < 8; ++r) {
      int row = r + half * 8;                 // 0..15 within tile
      int col = t * 16 + nrow;                // 0..79
      lds_B[wave * (16 * COLS) + row * COLS + col] = acc[t][r];
    }
  }
  __syncthreads();
  for (int e = tid; e < 16 * COLS; e += BDIM) {
    float s = 0.0f;
#pragma unroll
    for (int w = 0; w < 8; ++w) s += lds_B[w * (16 * COLS) + e];
    int row = e / COLS, col = e % COLS;
    atomicAdd(&S[((size_t)b * NPAD + ntile * 16 + row) * COLS + col], s);
  }
}

// ---------------- kernel 3: normalize, derive var/rm_cos, scatter ----------
__global__ void k_final(const float* __restrict__ S,
                        const float* __restrict__ qn,
                        const int* __restrict__ idx,
                        float* __restrict__ out) {
  int t = blockIdx.x * blockDim.x + threadIdx.x;
  if (t >= BS * NQ) return;
  int b = t / NQ, n = t % NQ;
  const float* s = S + ((size_t)b * NPAD + n) * COLS;
  float inv = 1.0f / (s[70] + 1e-9f);
  float qx = qn[n * 3 + 0], qy = qn[n * 3 + 1], qz = qn[n * 3 + 2];
  int g = idx[n];
  float* o = out + (size_t)b * COUT * HOWO + g;

  float m61 = 0.f, m62 = 0.f, m63 = 0.f;
  for (int c = 0; c < CFEA; ++c) {
    float mc = s[c] * inv;
    o[(size_t)(5 + c) * HOWO] = mc;
    if (c == 61) m61 = mc;
    if (c == 62) m62 = mc;
    if (c == 63) m63 = mc;
  }
  float var = (s[64] * inv - m61 * m61) + (s[65] * inv - m62 * m62) +
              (s[66] * inv - m63 * m63);
  float rm  = qx * (s[67] * inv) + qy * (s[68] * inv) + qz * (s[69] * inv);
  o[0]          = rm;
  o[HOWO]       = var;
  o[2 * HOWO]   = qx;
  o[3 * HOWO]   = qy;
  o[4 * HOWO]   = qz;
}

// ---------------- launcher ----------------
extern "C" void kernel_launch(void* const* d_in, const int* in_sizes, int n_in,
                              void* d_out, int out_size, void* d_ws, size_t ws_size,
                              hipStream_t stream) {
  const float* fea  = (const float*)d_in[0];
  const float* nrm  = (const float*)d_in[1];
  const float* msk  = (const float*)d_in[2];
  const float* lq   = (const float*)d_in[3];
  float* out = (float*)d_out;
  float* ws  = (float*)d_ws;

  float* qn   = ws + WS_QN;
  int*   idx  = (int*)(ws + WS_IDX);
  float* dmax = ws + WS_DMAX;
  float* S    = ws + WS_S;

  k_grid<<<1, 1, 0, stream>>>(qn, idx);
  k_zero<<<(BS * COUT * HOWO + 255) / 256, 256, 0, stream>>>(out, BS * COUT * HOWO);
  k_zero<<<(ZERO_CNT + 255) / 256, 256, 0, stream>>>(dmax, ZERO_CNT);
  k_max<<<dim3(NQ, BS), 256, 0, stream>>>(nrm, qn, dmax);
  k_main<<<dim3(NTILES, BS, KS), BDIM, 0, stream>>>(fea, nrm, msk, lq, qn, dmax, S);
  k_final<<<(BS * NQ + 127) / 128, 128, 0, stream>>>(S, qn, idx, out);
}